// TokenChoiceTopKRouter_28252294873408
// MI455X (gfx1250) — compile-verified
//
#include <hip/hip_runtime.h>
#include <hip/hip_bf16.h>

typedef __attribute__((ext_vector_type(2))) float v2f;
typedef __attribute__((ext_vector_type(8))) float v8f;
typedef unsigned int v4u __attribute__((ext_vector_type(4)));
typedef int v8i __attribute__((ext_vector_type(8)));
typedef int v4i __attribute__((ext_vector_type(4)));

#define DIM     4096
#define NEXP    64
#define TOPK    8
#define BK      32     // K-chunk
#define NCHUNK  (DIM / BK)
#define SR      36     // staged LDS row stride in floats (32 data + 4 pad) = 144 B
#define SSS     68     // score row stride (epilogue)

// smem float layout:
//  sB0 [0,2304)  sB1 [2304,4608)  sA0 [4608,6912)  sA1 [6912,9216)
//  sBias [9216,9280)  sHist [9280,9344)
#define OFF_SB0   0
#define OFF_SB1   2304
#define OFF_SA0   4608
#define OFF_SA1   6912
#define OFF_BIAS  9216
#define OFF_HIST  9280
#define SMEM_FLT  9344

// d_out layout (floats): [0,131072) top_scores, [131072,262144) indices (as float),
// [262144,262208) histogram.

__global__ void router_hist_init_kernel(float* __restrict__ hist) {
  hist[threadIdx.x] = 0.0f;
}

__device__ __forceinline__ unsigned lds_off(const void* p) {
  // LDS aperture: addr[31:0] is the LDS byte offset.
  return (unsigned)(unsigned long long)(uintptr_t)p;
}

// TDM: issue a 2D tensor-tile load (global -> LDS) via the Tensor Data Mover.
// g1 carries tensor/tile dims; g0 carries count/lds_addr/global_addr/type.
// 6-arg toolchain form: (g0 v4u, g1 v8i, g2 v4i, g3 v4i, extra v8i, cpol i32).
__device__ __forceinline__ void tdm_load(unsigned lds_byte_addr,
                                         unsigned long long gaddr,
                                         v8i g1) {
  v4u g0;
  g0.x = 1u;                                   // count=1 (valid), no gather/restore
  g0.y = lds_byte_addr;                        // [63:32] lds_addr
  g0.z = (unsigned)gaddr;                      // global_addr[31:0]
  g0.w = ((unsigned)(gaddr >> 32) & 0x01FFFFFFu) | (2u << 30);  // addr[56:32] | type=2
  v4i z4 = {0, 0, 0, 0};                       // groups 2/3 unused (2D tile)
  v8i z8 = {0, 0, 0, 0, 0, 0, 0, 0};
  __builtin_amdgcn_tensor_load_to_lds(g0, g1, z4, z4, z8, 0);
}

__global__ __launch_bounds__(128, 1)
void router_kernel(const float* __restrict__ x,
                   const float* __restrict__ w,
                   const float* __restrict__ bias,
                   float* __restrict__ out) {
  __shared__ float smem[SMEM_FLT];
  float* sBias = smem + OFF_BIAS;
  float* sHist = smem + OFF_HIST;

  const int tid  = threadIdx.x;
  const int wave = tid >> 5;        // 0..3
  const int lane = tid & 31;
  const int half = lane >> 4;       // 0 or 1
  const int l16  = lane & 15;

  if (tid < NEXP) { sBias[tid] = bias[tid]; sHist[tid] = 0.0f; }

  const int tokBase = blockIdx.x * 64 + wave * 16;   // 256 blocks x 64 tokens
  const unsigned long long xbase = (unsigned long long)(uintptr_t)x;
  const unsigned long long wbase = (unsigned long long)(uintptr_t)w;

  // D# group1: data_size=4B(2), pad_enable, pad_interval=4 (32 DWORDs = 1 row),
  // pad_amount=3 (4 DWORDs) -> LDS row stride 36 floats. tile = 32 x 16.
  const unsigned g1w0 = (2u << 16) | (1u << 20) | (4u << 22) | (3u << 25);
  const v8i g1x = { (int)g1w0, (int)(4096u << 16), (int)(16384u << 16),
                    (int)(32u << 16), 16, 4096, 0, 0 };   // x: 16384 x 4096
  const v8i g1g = { (int)g1w0, (int)(4096u << 16), (int)(64u << 16),
                    (int)(32u << 16), 16, 4096, 0, 0 };   // gate: 64 x 4096

  // Per-wave LDS destinations (each wave stages its x tile + 16 gate rows).
  const unsigned ldsA[2] = { (unsigned)((OFF_SA0 + wave * 16 * SR) * 4),
                             (unsigned)((OFF_SA1 + wave * 16 * SR) * 4) };
  const unsigned ldsB[2] = { (unsigned)((OFF_SB0 + wave * 16 * SR) * 4),
                             (unsigned)((OFF_SB1 + wave * 16 * SR) * 4) };
  const unsigned lbase = lds_off(smem);

  v8f c0 = {}, c1 = {}, c2 = {}, c3 = {};

  // Prologue: kick off chunk 0 into buffer 0.
  tdm_load(lbase + ldsA[0], xbase + ((unsigned long long)tokBase * DIM) * 4ull, g1x);
  tdm_load(lbase + ldsB[0], wbase + ((unsigned long long)(wave * 16) * DIM) * 4ull, g1g);

  for (int i = 0; i < NCHUNK; ++i) {
    const int cb = i & 1;
    if (i + 1 < NCHUNK) {
      const int nb = (i + 1) & 1;
      const unsigned long long koff = (unsigned long long)((i + 1) * BK) * 4ull;
      tdm_load(lbase + ldsA[nb],
               xbase + ((unsigned long long)tokBase * DIM) * 4ull + koff, g1x);
      tdm_load(lbase + ldsB[nb],
               wbase + ((unsigned long long)(wave * 16) * DIM) * 4ull + koff, g1g);
      __builtin_amdgcn_s_wait_tensorcnt((short)2);  // this chunk's 2 TDMs done
    } else {
      __builtin_amdgcn_s_wait_tensorcnt((short)0);
    }
    __syncthreads();   // all waves' slices (incl. gate rows) landed in LDS

    const float* sAw = smem + (cb ? OFF_SA1 : OFF_SA0) + wave * 16 * SR;
    const float* sBc = smem + (cb ? OFF_SB1 : OFF_SB0);
#pragma unroll
    for (int kk = 0; kk < BK; kk += 4) {
      v2f a  = *reinterpret_cast<const v2f*>(sAw + l16 * SR + kk + half * 2);
      v2f b0 = *reinterpret_cast<const v2f*>(sBc + (l16)      * SR + kk + half * 2);
      v2f b1 = *reinterpret_cast<const v2f*>(sBc + (16 + l16) * SR + kk + half * 2);
      v2f b2 = *reinterpret_cast<const v2f*>(sBc + (32 + l16) * SR + kk + half * 2);
      v2f b3 = *reinterpret_cast<const v2f*>(sBc + (48 + l16) * SR + kk + half * 2);
      c0 = __builtin_amdgcn_wmma_f32_16x16x4_f32(false, a, false, b0, (short)0, c0, false, false);
      c1 = __builtin_amdgcn_wmma_f32_16x16x4_f32(false, a, false, b1, (short)0, c1, false, false);
      c2 = __builtin_amdgcn_wmma_f32_16x16x4_f32(false, a, false, b2, (short)0, c2, false, false);
      c3 = __builtin_amdgcn_wmma_f32_16x16x4_f32(false, a, false, b3, (short)0, c3, false, false);
    }
    __syncthreads();   // safe to overwrite the other buffer next iteration
  }

  // Dump sigmoid(logits) to LDS (buffers are dead now).
  // C layout: VGPR r -> token r (lanes 0-15) / token 8+r (lanes 16-31); N = l16.
  float* sS = smem + wave * 16 * SSS;
#pragma unroll
  for (int r = 0; r < 8; ++r) {
    int trow = r + 8 * half;
    sS[trow * SSS +  0 + l16] = 1.0f / (1.0f + __expf(-c0[r]));
    sS[trow * SSS + 16 + l16] = 1.0f / (1.0f + __expf(-c1[r]));
    sS[trow * SSS + 32 + l16] = 1.0f / (1.0f + __expf(-c2[r]));
    sS[trow * SSS + 48 + l16] = 1.0f / (1.0f + __expf(-c3[r]));
  }
  __syncthreads();

  if (lane < 16) {
    const int tl = l16;
    const int t  = tokBase + l16;
    float sc[NEXP];
#pragma unroll
    for (int e = 0; e < NEXP; ++e) sc[e] = sS[tl * SSS + e] + sBias[e];

    unsigned long long mask = 0ull;
    float tops[TOPK];
    int   tix[TOPK];
    float ssum = 0.0f;
#pragma unroll
    for (int i = 0; i < TOPK; ++i) {
      float best = -3.0e38f;
      int   bi   = 0;
#pragma unroll
      for (int e = 0; e < NEXP; ++e) {
        bool used = (mask >> e) & 1ull;
        if (!used && sc[e] > best) { best = sc[e]; bi = e; }
      }
      mask |= (1ull << bi);
      float s = sS[tl * SSS + bi];   // unbiased sigmoid score
      tops[i] = s;
      tix[i]  = bi;
      ssum   += s;
    }
    const float scale = 2.5f / (ssum + 1e-20f);
#pragma unroll
    for (int i = 0; i < TOPK; ++i) {
      out[(size_t)t * TOPK + i]          = tops[i] * scale;
      out[131072 + (size_t)t * TOPK + i] = (float)tix[i];
      atomicAdd(&sHist[tix[i]], 1.0f);
    }
  }
  __syncthreads();
  if (tid < NEXP) atomicAdd(out + 262144 + tid, sHist[tid]);
}

extern "C" void kernel_launch(void* const* d_in, const int* in_sizes, int n_in,
                              void* d_out, int out_size, void* d_ws, size_t ws_size,
                              hipStream_t stream) {
  const float* x    = (const float*)d_in[0];  // [16384, 4096]
  const float* w    = (const float*)d_in[1];  // [64, 4096]
  const float* bias = (const float*)d_in[2];  // [64]
  float* out = (float*)d_out;

  router_hist_init_kernel<<<1, NEXP, 0, stream>>>(out + 262144);
  router_kernel<<<256, 128, 0, stream>>>(x, w, bias, out);
}